// RelationAwareSelfAttention_42288247996698
// MI455X (gfx1250) — compile-verified
//
#include <hip/hip_runtime.h>
#include <math.h>

typedef __attribute__((ext_vector_type(2))) float v2f;
typedef __attribute__((ext_vector_type(8))) float v8f;
typedef int v4i __attribute__((vector_size(16)));          // matches builtin param
typedef __attribute__((address_space(1))) v4i gv4i;        // global int4
typedef __attribute__((address_space(3))) v4i lv4i;        // LDS int4

#define NTOK   500
#define DMODEL 2048
#define KWIN   10
#define NREL   (2 * KWIN + 1)

#define KB    16    // K-slab per LDS stage
#define BSTR  272   // Bs row stride in floats (256 + 16): conflict-free, 16B aligned
#define ASTR  20    // As row stride in floats: conflict-free, 16B aligned

#ifndef __has_builtin
#define __has_builtin(x) 0
#endif
#if __has_builtin(__builtin_amdgcn_global_load_async_to_lds_b128)
#define ASYNC_LDS 1
#else
#define ASYNC_LDS 0
#endif

__device__ __forceinline__ int imin(int a, int b) { return a < b ? a : b; }
__device__ __forceinline__ int imax(int a, int b) { return a > b ? a : b; }

// 16-byte global -> LDS stage. Uses the CDNA5 async-to-LDS path when the
// toolchain exposes it (tracked by ASYNCcnt), else a plain b128 load + ds store.
__device__ __forceinline__ void stage16(const float* __restrict__ src, float* dst)
{
#if ASYNC_LDS
    // Low 32 bits of a generic LDS address are the LDS byte offset (ISA 10.2).
    __builtin_amdgcn_global_load_async_to_lds_b128(
        (gv4i*)(uintptr_t)src, (lv4i*)(uint32_t)(uintptr_t)dst, 0, 0);
#else
    *(float4*)dst = *(const float4*)src;
#endif
}

__device__ __forceinline__ void stage_fence()
{
#if ASYNC_LDS
#if __has_builtin(__builtin_amdgcn_s_wait_asynccnt)
    __builtin_amdgcn_s_wait_asynccnt(0);
#else
    asm volatile("s_wait_asynccnt 0x0" ::: "memory");
#endif
#endif
    __syncthreads();
}

// ---------------------------------------------------------------------------
// NN GEMM, LDS-staged + double-buffered:  C[M,N] = alpha * A[M,K] * B[K,N]
// Requires N % 256 == 0 (true for every NN GEMM here: N = 2048).
// Handles arbitrary K (zero-fills the slab tail; needed for K = 500).
// Block = 256 threads = 8 waves; block tile 32(M) x 256(N); wave tile 16 x 64.
// Fragments per V_WMMA_F32_16X16X4_F32 layout:
//   A frag: lane&15 = M row; half = lane>>4 -> K pair {0,1}/{2,3}  (ds_load_b64)
//   B frag: lane&15 = N col; vgpr = K within pair; half = K block  (2x ds_load_b32)
// ---------------------------------------------------------------------------
__global__ __launch_bounds__(256) void gemm_nn_wmma(
    const float* __restrict__ A, const float* __restrict__ B,
    float* __restrict__ C, int M, int N, int Kd,
    int lda, int ldb, int ldc, float alpha)
{
    __shared__ float Bs[2][KB * BSTR];
    __shared__ float As[2][32 * ASTR];

    const int tid  = threadIdx.x;
    const int lane = tid & 31;
    const int wave = tid >> 5;
    const int half = lane >> 4;
    const int l16  = lane & 15;
    const int wrow = wave & 1;   // 2 wave-rows of 16
    const int wcol = wave >> 1;  // 4 wave-cols of 64

    const int row0b = blockIdx.y * 32;
    const int col0b = blockIdx.x * 256;

    // -- staging assignments (per thread) --
    // B slab: KB x 256 floats = 1024 vec4 -> 4 vec4/thread
    // A slab: 32 x KB floats  = 128 vec4  -> threads 0..127, 1 vec4 each
    const int arow = tid >> 2;          // 0..63 (use 0..31)
    const int ac4  = (tid & 3) << 2;    // 0,4,8,12

    v8f acc[4] = {};

    const int nk = (Kd + KB - 1) / KB;

    // ---- stage slab kt into buffer buf ----
    auto stage = [&](int kt, int buf) {
        const int k = kt * KB;
#pragma unroll
        for (int i = 0; i < 4; ++i) {
            const int v  = tid + 256 * i;
            const int r  = v >> 6;          // 0..15
            const int c4 = (v & 63) << 2;   // 0..252
            float* dst = &Bs[buf][r * BSTR + c4];
            const int gk = k + r;
            if (gk < Kd) {
                stage16(B + (size_t)gk * ldb + (col0b + c4), dst);
            } else {
                dst[0] = 0.f; dst[1] = 0.f; dst[2] = 0.f; dst[3] = 0.f;
            }
        }
        if (tid < 128) {
            float* dst = &As[buf][arow * ASTR + ac4];
            const int grow = imin(row0b + arow, M - 1);
            const int gk = k + ac4;
            const float* src = A + (size_t)grow * lda + gk;
            if (gk + 3 < Kd) {
                stage16(src, dst);
            } else {
#pragma unroll
                for (int e = 0; e < 4; ++e)
                    dst[e] = (gk + e < Kd) ? src[e] : 0.f;
            }
        }
    };

    stage(0, 0);
    stage_fence();

    for (int kt = 0; kt < nk; ++kt) {
        const int buf = kt & 1;
        if (kt + 1 < nk) stage(kt + 1, buf ^ 1);  // overlap next slab (ASYNCcnt)

        const float* as = &As[buf][(wrow * 16 + l16) * ASTR + 2 * half];
        const float* bs = &Bs[buf][(2 * half) * BSTR + wcol * 64 + l16];

#pragma unroll
        for (int ks = 0; ks < KB; ks += 4) {
            v2f a = *(const v2f*)(as + ks);   // ds_load_b64, conflict-free
#pragma unroll
            for (int t = 0; t < 4; ++t) {
                v2f b;
                b.x = bs[ks * BSTR + 16 * t];
                b.y = bs[ks * BSTR + BSTR + 16 * t];
                acc[t] = __builtin_amdgcn_wmma_f32_16x16x4_f32(
                    false, a, false, b, (short)0, acc[t], false, false);
            }
        }
        stage_fence();  // next buffer ready; current buffer safe to overwrite
    }

#pragma unroll
    for (int t = 0; t < 4; ++t) {
        const int col = col0b + wcol * 64 + 16 * t + l16;
        if (col >= N) continue;
#pragma unroll
        for (int r = 0; r < 8; ++r) {
            const int row = row0b + wrow * 16 + r + 8 * half;
            if (row < M)
                C[(size_t)row * ldc + col] = alpha * acc[t][r];
        }
    }
}

// ---------------------------------------------------------------------------
// NT GEMM (direct from global; per-lane loads are contiguous float2):
//   C[M,N] = A[M,K] * B^T where B is row-major [N,K].  Used for S = Qs @ K^T.
// Pointer-bumped induction: no 64-bit multiplies in the loop.
// ---------------------------------------------------------------------------
__global__ __launch_bounds__(256) void gemm_nt_wmma(
    const float* __restrict__ A, const float* __restrict__ B,
    float* __restrict__ C, int M, int N, int Kd,
    int lda, int ldb, int ldc, float alpha)
{
    const int lane = threadIdx.x & 31;
    const int wave = threadIdx.x >> 5;
    const int half = lane >> 4;
    const int l16  = lane & 15;

    const int row0 = blockIdx.y * 32 + (wave & 1) * 16;
    const int col0 = blockIdx.x * 256 + (wave >> 1) * 64;

    v8f acc[4] = {};

    const float* ap = A + (size_t)imin(row0 + l16, M - 1) * lda + 2 * half;
    const float* bp[4];
#pragma unroll
    for (int t = 0; t < 4; ++t)
        bp[t] = B + (size_t)imin(col0 + 16 * t + l16, N - 1) * ldb + 2 * half;

    for (int k = 0; k < Kd; k += 4) {
        v2f a = *(const v2f*)ap; ap += 4;
#pragma unroll
        for (int t = 0; t < 4; ++t) {
            v2f b = *(const v2f*)bp[t]; bp[t] += 4;
            acc[t] = __builtin_amdgcn_wmma_f32_16x16x4_f32(
                false, a, false, b, (short)0, acc[t], false, false);
        }
    }

#pragma unroll
    for (int t = 0; t < 4; ++t) {
        const int col = col0 + 16 * t + l16;
        if (col >= N) continue;
#pragma unroll
        for (int r = 0; r < 8; ++r) {
            const int row = row0 + r + 8 * half;
            if (row < M)
                C[(size_t)row * ldc + col] = alpha * acc[t][r];
        }
    }
}

// ---------------------------------------------------------------------------
// QR[i, r] = Qs[i, :] . wk[r, :]   (500 x 21, tiny)
// ---------------------------------------------------------------------------
__global__ __launch_bounds__(256) void qr_kernel(
    const float* __restrict__ Qs, const float* __restrict__ wkTab,
    float* __restrict__ QR)
{
    __shared__ float red[256];
    const int i = blockIdx.x;
    const int t = threadIdx.x;
    const float* q = Qs + (size_t)i * DMODEL;
    for (int r = 0; r < NREL; ++r) {
        const float* w = wkTab + (size_t)r * DMODEL;
        float p = 0.f;
        for (int k = t; k < DMODEL; k += 256) p += q[k] * w[k];
        red[t] = p;
        __syncthreads();
        for (int s = 128; s > 0; s >>= 1) {
            if (t < s) red[t] += red[t + s];
            __syncthreads();
        }
        if (t == 0) QR[i * NREL + r] = red[0];
        __syncthreads();
    }
}

// ---------------------------------------------------------------------------
// Fused epilogue per row: rel-pos bias gather + mask + softmax -> A
// ---------------------------------------------------------------------------
__global__ __launch_bounds__(256) void bias_mask_softmax(
    const float* __restrict__ S, const float* __restrict__ QR,
    const int* __restrict__ mask, float* __restrict__ Aout)
{
    __shared__ float red[256];
    const int i = blockIdx.x;
    const int t = threadIdx.x;
    const float* Srow = S + (size_t)i * NTOK;
    const int* Mrow = mask + (size_t)i * NTOK;
    const float* QRrow = QR + i * NREL;

    float v0 = -3.0e38f, v1 = -3.0e38f;
    if (t < NTOK) {
        const int d = imin(imax(t - i, -KWIN), KWIN);
        float s = Srow[t] + QRrow[d + KWIN];
        v0 = (Mrow[t] == 0) ? -1e9f : s;
    }
    const int j1 = t + 256;
    if (j1 < NTOK) {
        const int d = imin(imax(j1 - i, -KWIN), KWIN);
        float s = Srow[j1] + QRrow[d + KWIN];
        v1 = (Mrow[j1] == 0) ? -1e9f : s;
    }

    red[t] = fmaxf(v0, v1);
    __syncthreads();
    for (int s = 128; s > 0; s >>= 1) {
        if (t < s) red[t] = fmaxf(red[t], red[t + s]);
        __syncthreads();
    }
    const float rowmax = red[0];
    __syncthreads();

    const float e0 = (t < NTOK) ? expf(v0 - rowmax) : 0.f;
    const float e1 = (j1 < NTOK) ? expf(v1 - rowmax) : 0.f;
    red[t] = e0 + e1;
    __syncthreads();
    for (int s = 128; s > 0; s >>= 1) {
        if (t < s) red[t] += red[t + s];
        __syncthreads();
    }
    const float inv = 1.0f / red[0];

    float* Arow = Aout + (size_t)i * NTOK;
    if (t < NTOK) Arow[t] = e0 * inv;
    if (j1 < NTOK) Arow[j1] = e1 * inv;
}

// ---------------------------------------------------------------------------
extern "C" void kernel_launch(void* const* d_in, const int* in_sizes, int n_in,
                              void* d_out, int out_size, void* d_ws, size_t ws_size,
                              hipStream_t stream)
{
    const float* X     = (const float*)d_in[0];   // [500, 2048]
    const int*   mask  = (const int*)  d_in[1];   // [500, 500]
    const float* Wq    = (const float*)d_in[2];   // [2048, 2048]
    const float* Wk    = (const float*)d_in[3];
    const float* Wv    = (const float*)d_in[4];
    const float* Wo    = (const float*)d_in[5];
    const float* wkTab = (const float*)d_in[6];   // [21, 2048]

    const size_t sz_nm = (size_t)NTOK * DMODEL;   // 1,024,000 floats
    float* ws = (float*)d_ws;
    float* Qs = ws;                                // [500,2048] (pre-scaled)
    float* Km = Qs + sz_nm;                        // [500,2048]
    float* Vm = Km + sz_nm;                        // [500,2048]
    float* S  = Vm + sz_nm;                        // [500,500]
    float* QR = S + (size_t)NTOK * NTOK;           // [500,21]
    float* Y1 = Qs;                                // reuse: Qs dead after score GEMM

    float* Y = (float*)d_out;                      // [500,2048]
    float* A = Y + sz_nm;                          // [500,500]

    const float scale = 1.0f / sqrtf((float)DMODEL);

    dim3 blk(256);
    dim3 gProj((DMODEL + 255) / 256, (NTOK + 31) / 32);   // N=2048 GEMMs
    dim3 gScore((NTOK + 255) / 256, (NTOK + 31) / 32);    // N=500 GEMM

    // Q/K/V projections (scale folded into Q).
    gemm_nn_wmma<<<gProj, blk, 0, stream>>>(X, Wq, Qs, NTOK, DMODEL, DMODEL, DMODEL, DMODEL, DMODEL, scale);
    gemm_nn_wmma<<<gProj, blk, 0, stream>>>(X, Wk, Km, NTOK, DMODEL, DMODEL, DMODEL, DMODEL, DMODEL, 1.0f);
    gemm_nn_wmma<<<gProj, blk, 0, stream>>>(X, Wv, Vm, NTOK, DMODEL, DMODEL, DMODEL, DMODEL, DMODEL, 1.0f);

    // Relative-position projections QR = Qs @ wk^T.
    qr_kernel<<<NTOK, blk, 0, stream>>>(Qs, wkTab, QR);

    // Content scores S = Qs @ K^T.
    gemm_nt_wmma<<<gScore, blk, 0, stream>>>(Qs, Km, S, NTOK, NTOK, DMODEL, DMODEL, DMODEL, NTOK, 1.0f);

    // Bias gather + mask + softmax -> A (second output).
    bias_mask_softmax<<<NTOK, blk, 0, stream>>>(S, QR, mask, A);

    // Y1 = A @ V   (K = 500: zero-filled slab tail)
    gemm_nn_wmma<<<gProj, blk, 0, stream>>>(A, Vm, Y1, NTOK, DMODEL, NTOK, NTOK, DMODEL, DMODEL, 1.0f);

    // Y = Y1 @ Wo  (first output)
    gemm_nn_wmma<<<gProj, blk, 0, stream>>>(Y1, Wo, Y, NTOK, DMODEL, DMODEL, DMODEL, DMODEL, DMODEL, 1.0f);
}